// conv2_59700045414487
// MI455X (gfx1250) — compile-verified
//
#include <hip/hip_runtime.h>

typedef __attribute__((ext_vector_type(2))) float v2f;
typedef __attribute__((ext_vector_type(8))) float v8f;

// Problem constants from the reference (fixed shapes).
#define LAT 32           // lattice side
#define SITES 1024       // LAT*LAT
#define CIN 4            // input channels (complex)
#define FOUT 8           // output features (complex)
#define BATCH 64
// Real GEMM view: M = BATCH*SITES = 65536, K = 32, N = 16.
// One wave handles one 16-row M-tile via 8x V_WMMA_F32_16X16X4_F32.

__global__ __launch_bounds__(256) void conv2_wmma_f32(
    const float* __restrict__ x,    // (B,S,C) complex64 -> interleaved floats
    const float* __restrict__ W,    // (4,4,8) complex64 -> 256 floats
    const float* __restrict__ bias, // (8,)    complex64 -> 16 floats
    float* __restrict__ out)        // (B,S,F) complex64 -> interleaved floats
{
  const int lane = threadIdx.x & 31;
  const int wave = threadIdx.x >> 5;
  const int gwave = blockIdx.x * (blockDim.x >> 5) + wave;
  const int mbase = gwave << 4;          // 16 M-rows per tile

  const int l = lane & 15;               // row-in-tile for A; column N for B/C/D
  const int h = lane >> 4;               // lane half selects K sub-pair

  // ---------------- B fragments: W' is (K=32) x (N=16) real expansion of W.
  // K index = tap*8 + chan*2 + q (q: 0=re,1=im of x). N index = 2f + p.
  // q=0 row: [p==0 -> Wre, p==1 -> Wim];  q=1 row: [p==0 -> -Wim, p==1 -> Wre].
  // Step t covers K=[4t,4t+4): this lane needs (tap=t>>1, chan=2*(t&1)+h, q=0,1).
  const int f = l >> 1;
  const int p = l & 1;
  v2f breg[8];
#pragma unroll
  for (int t = 0; t < 8; ++t) {
    const int k = t >> 1;
    const int c = ((t & 1) << 1) + h;
    const float2 w = *(const float2*)(W + ((((k << 2) + c) << 3) + f) * 2);
    breg[t].x = p ? w.y : w.x;           // q=0 row of this K-pair
    breg[t].y = p ? w.x : -w.y;          // q=1 row of this K-pair
  }

  // ---------------- C init = bias (depends only on column N = l).
  const float bv = bias[l];
  v8f acc;
#pragma unroll
  for (int g = 0; g < 8; ++g) acc[g] = bv;

  // ---------------- Gather A fragments: analytic periodic 2x2 neighborhood.
  // Row m = mbase + l  ->  (b, j);  tap k=(dr,dc): i = ((r+dr)%32)*32 + (c+dc)%32.
  // Each tap's site data is 8 contiguous floats (4 complex channels = 32B).
  // Lane half h takes float pairs (2h,2h+1) and (4+2h,4+2h+1) of each tap,
  // so the wave reads each 32B chunk exactly once.
  const int m = mbase + l;
  const int bidx = m >> 10;
  const int j = m & (SITES - 1);
  const int r = j >> 5, cc = j & (LAT - 1);
  v2f areg[8];
#pragma unroll
  for (int k = 0; k < 4; ++k) {
    const int dr = k >> 1, dc = k & 1;
    const int i = (((r + dr) & (LAT - 1)) << 5) + ((cc + dc) & (LAT - 1));
    const float* src = x + (((size_t)(bidx << 10) + (size_t)i) << 3) + (h << 1);
    areg[2 * k]     = *(const v2f*)(src);      // K = 8k + 2h, 8k + 2h + 1
    areg[2 * k + 1] = *(const v2f*)(src + 4);  // K = 8k + 4 + 2h, ... + 1
  }

  // ---------------- 8 chained WMMA steps: D = A(16x4) * B(4x16) + C.
#pragma unroll
  for (int t = 0; t < 8; ++t) {
    acc = __builtin_amdgcn_wmma_f32_16x16x4_f32(
        /*neg_a=*/false, areg[t], /*neg_b=*/false, breg[t],
        /*c_mod=*/(short)0, acc, /*reuse_a=*/false, /*reuse_b=*/false);
  }

  // ---------------- Store. D layout: VGPR g -> row mbase+g (lanes 0-15) /
  // mbase+8+g (lanes 16-31), col = l. Each store writes contiguous 64B chunks.
  float* obase = out + (((size_t)(mbase + (h << 3))) << 4) + l;
#pragma unroll
  for (int g = 0; g < 8; ++g) {
    obase[(size_t)g << 4] = acc[g];
  }
}

extern "C" void kernel_launch(void* const* d_in, const int* in_sizes, int n_in,
                              void* d_out, int out_size, void* d_ws, size_t ws_size,
                              hipStream_t stream) {
  (void)in_sizes; (void)n_in; (void)d_ws; (void)ws_size; (void)out_size;
  const float* x    = (const float*)d_in[0];   // complex64 viewed as float pairs
  const float* W    = (const float*)d_in[1];
  const float* bias = (const float*)d_in[2];
  // d_in[3] = kernel2 index map: provably reconstructible analytically; unused.
  float* out = (float*)d_out;

  // 65536 M-rows / 16 per wave-tile = 4096 waves; 8 waves per 256-thread block.
  const int tiles = (BATCH * SITES) / 16;      // 4096
  const int wavesPerBlock = 256 / 32;          // 8
  dim3 grid(tiles / wavesPerBlock);            // 512
  dim3 block(256);
  conv2_wmma_f32<<<grid, block, 0, stream>>>(x, W, bias, out);
}